// _ProposalLayer_83906481095059
// MI455X (gfx1250) — compile-verified
//
#include <hip/hip_runtime.h>

typedef unsigned int       u32;
typedef unsigned long long u64;
typedef unsigned int v4u __attribute__((ext_vector_type(4)));
typedef int          v8i __attribute__((ext_vector_type(8)));
typedef int          v4i __attribute__((ext_vector_type(4)));

#define TOTAL   147456          // 9*128*128
#define NPAD1   262144          // 2^18
#define K       6000
#define NPAD2   8192            // 2^13
#define NW      188             // ceil(6000/32)
#define ST      6016            // padded column stride (16B aligned rows)
#define TC      1024            // TDM tile columns
#define NMS_T   0.7f

// ---- workspace layout (bytes) ----
#define OFF_BOXES   0u          // float4 * 147456
#define OFF_KEYS1   2359296u    // u64 * 262144
#define OFF_PROPS   4456448u    // float4 * 6000 (top-k order)
#define OFF_KEYS2   4552448u    // u64 * 8192
#define OFF_SPROPS  4617984u    // float4 * 6000 (y2-sorted order)
#define OFF_COLS    4713984u    // float * 5*6016  (x1,y1,x2,y2,area SoA)
#define OFF_MASK    4834304u    // u32 * 6000*188
#define OFF_VALID   9346304u    // u32 * 188
#define OFF_PREFIX  9347072u    // u32 * 188

__device__ __constant__ float c_sizes[9] = {4.f,8.f,12.f,16.f,24.f,32.f,48.f,64.f,96.f};

__device__ __forceinline__ u32 ord_f32(float f) {
  u32 u = __float_as_uint(f);
  return u ^ ((u >> 31) ? 0xFFFFFFFFu : 0x80000000u);
}

// ---- stage 1: decode anchors+deltas, clip, min-size mask, pack sort keys ----
__global__ __launch_bounds__(256) void build_boxes(const float* __restrict__ scores,
                                                   const float* __restrict__ deltas,
                                                   float4* __restrict__ boxes,
                                                   u64* __restrict__ keys) {
  int i = blockIdx.x * blockDim.x + threadIdx.x;
  if (i >= NPAD1) return;
  if (i >= TOTAL) { keys[i] = 0ull; return; }
  int a = i >> 14;
  int hw = i & 16383;
  int h = hw >> 7, w = hw & 127;
  float s = c_sizes[a];
  float4 d = ((const float4*)deltas)[i];           // reshape(9,128,128,4) == linear float4
  float x0 = (h + 0.5f) - 0.5f * s + d.x;
  float y0 = (w + 0.5f) - 0.5f * s + d.y;
  float bw = s + d.z;
  float bh = s + d.w;
  x0 = fmaxf(x0, 0.f); y0 = fmaxf(y0, 0.f); bw = fmaxf(bw, 0.f); bh = fmaxf(bh, 0.f);
  float x1 = fminf(x0, 128.f), y1 = fminf(y0, 128.f);
  float x2 = fminf(x0 + bw, 128.f), y2 = fminf(y0 + bh, 128.f);
  float ww = x2 - x1, hh = y2 - y1;
  boxes[i] = make_float4(x1, y1, ww, hh);
  bool keep = (ww >= 3.0f) && (hh >= 3.0f);
  float sc = keep ? scores[i] : -__builtin_huge_valf();
  keys[i] = ((u64)ord_f32(sc) << 32) | (u64)(0xFFFFFFFFu - (u32)i);
}

// ---- global bitonic step (descending; key packs stable tiebreak), j >= 512 ----
__global__ __launch_bounds__(256) void bitonic_step(u64* __restrict__ keys, u32 j, u32 k, u32 n) {
  u32 i = blockIdx.x * blockDim.x + threadIdx.x;
  if (i >= n) return;
  u32 ixj = i ^ j;
  if (ixj > i) {
    u64 a = keys[i], b = keys[ixj];
    bool up = ((i & k) == 0);
    if (up ? (a < b) : (a > b)) { keys[i] = b; keys[ixj] = a; }
  }
}

// ---- fused LDS bitonic: all outer stages k=2..512 in one launch ----
__global__ __launch_bounds__(256) void bitonic_local_full(u64* __restrict__ keys) {
  __shared__ u64 sh[512];
  u32 base = blockIdx.x * 512u;
  u32 t = threadIdx.x;
  sh[t] = keys[base + t];
  sh[t + 256] = keys[base + t + 256];
  __syncthreads();
  for (u32 kk = 2; kk <= 512; kk <<= 1) {
    for (u32 j = kk >> 1; j > 0; j >>= 1) {
      u32 i = ((t & ~(j - 1u)) << 1) | (t & (j - 1u));   // insert 0 at bit log2(j)
      u32 ixj = i | j;
      bool up = (((base + i) & kk) == 0);
      u64 a = sh[i], b = sh[ixj];
      if (up ? (a < b) : (a > b)) { sh[i] = b; sh[ixj] = a; }
      __syncthreads();
    }
  }
  keys[base + t] = sh[t];
  keys[base + t + 256] = sh[t + 256];
}

// ---- fused LDS bitonic: tail steps j=256..1 of outer stage k in one launch ----
__global__ __launch_bounds__(256) void bitonic_local(u64* __restrict__ keys, u32 k) {
  __shared__ u64 sh[512];
  u32 base = blockIdx.x * 512u;
  u32 t = threadIdx.x;
  sh[t] = keys[base + t];
  sh[t + 256] = keys[base + t + 256];
  __syncthreads();
  for (u32 j = 256; j > 0; j >>= 1) {
    u32 i = ((t & ~(j - 1u)) << 1) | (t & (j - 1u));
    u32 ixj = i | j;
    bool up = (((base + i) & k) == 0);
    u64 a = sh[i], b = sh[ixj];
    if (up ? (a < b) : (a > b)) { sh[i] = b; sh[ixj] = a; }
    __syncthreads();
  }
  keys[base + t] = sh[t];
  keys[base + t + 256] = sh[t + 256];
}

// ---- stage 3: gather top-6000 props, build y2-sort keys ----
__global__ __launch_bounds__(256) void gather_props(const u64* __restrict__ keys1,
                                                    const float4* __restrict__ boxes,
                                                    float4* __restrict__ props,
                                                    u64* __restrict__ keys2) {
  int t = blockIdx.x * blockDim.x + threadIdx.x;
  if (t >= NPAD2) return;
  if (t >= K) { keys2[t] = 0ull; return; }
  u64 kk = keys1[t];
  u32 idx = 0xFFFFFFFFu - (u32)(kk & 0xFFFFFFFFu);
  float4 p = boxes[idx];
  props[t] = p;
  float y2 = p.y + p.w;
  keys2[t] = ((u64)ord_f32(y2) << 32) | (u64)(0xFFFFFFFFu - (u32)t);
}

// ---- stage 5: build y2-sorted SoA columns + areas ----
__global__ __launch_bounds__(256) void build_cols(const u64* __restrict__ keys2,
                                                  const float4* __restrict__ props,
                                                  float4* __restrict__ sprops,
                                                  float* __restrict__ cols) {
  int t = blockIdx.x * blockDim.x + threadIdx.x;
  if (t >= K) return;
  u64 kk = keys2[t];
  u32 src = 0xFFFFFFFFu - (u32)(kk & 0xFFFFFFFFu);
  float4 p = props[src];
  sprops[t] = p;
  float x1 = p.x, y1 = p.y, x2 = x1 + p.z, y2 = y1 + p.w;
  float area = fmaxf((x2 - x1) * (y2 - y1), 1e-6f);
  cols[0 * ST + t] = x1;
  cols[1 * ST + t] = y1;
  cols[2 * ST + t] = x2;
  cols[3 * ST + t] = y2;
  cols[4 * ST + t] = area;
}

// ---- stage 6: 6000x6000 suppression bitmask; TDM stages column tiles to LDS ----
__global__ __launch_bounds__(256) void nms_mask(const float* __restrict__ cols,
                                                u32* __restrict__ mask) {
  __shared__ float tile[5 * TC];                    // 20 KB, LDS offset 0
  int i = blockIdx.x;                               // row index
  float rx1 = cols[0 * ST + i], ry1 = cols[1 * ST + i];
  float rx2 = cols[2 * ST + i], ry2 = cols[3 * ST + i];
  int lane = threadIdx.x & 31;
  int wave = threadIdx.x >> 5;

  for (int ct = 0; ct < 6; ++ct) {
    int c0 = ct * TC;
    if (threadIdx.x < 32) {                         // wave 0 issues the TDM op (EXEC ignored)
      u64 gaddr = (u64)(const void*)(cols + c0);
      u32 tdim0 = (u32)(K - c0);                    // remaining valid cols; OOB reads -> 0
      v4u g0;
      g0.x = 1u;                                    // count=1, user descriptor
      g0.y = 0u;                                    // lds_addr = 0 (tile[] is first LDS var)
      g0.z = (u32)(gaddr & 0xFFFFFFFFu);
      g0.w = (u32)((gaddr >> 32) & 0x01FFFFFFu) | (2u << 30);  // type=2 ("image")
      v8i g1;
      g1[0] = (int)0x00020000u;                     // data_size=4B, no flags
      g1[1] = (int)((tdim0 & 0xFFFFu) << 16);       // tensor_dim0 lo16
      g1[2] = (int)(((tdim0 >> 16) & 0xFFFFu) | (5u << 16));   // tensor_dim0 hi16 | tensor_dim1 lo16
      g1[3] = (int)(0u | ((u32)TC << 16));          // tensor_dim1 hi16 | tile_dim0=1024
      g1[4] = (int)(5u);                            // tile_dim1=5, tile_dim2=0
      g1[5] = (int)((u32)ST);                       // tensor_dim0_stride lo32 = 6016 elements
      g1[6] = 0;                                    // stride0 hi16 | stride1 lo16
      g1[7] = 0;
      v4i gz4 = {0, 0, 0, 0};
      v8i gz8 = {0, 0, 0, 0, 0, 0, 0, 0};
      __builtin_amdgcn_tensor_load_to_lds(g0, g1, gz4, gz4, gz8, 0);
      __builtin_amdgcn_s_wait_tensorcnt(0);
    }
    __syncthreads();

    for (int widx = wave; widx < TC / 32; widx += 8) {
      int li = widx * 32 + lane;
      int j = c0 + li;
      float cx1 = tile[0 * TC + li];
      float cy1 = tile[1 * TC + li];
      float cx2 = tile[2 * TC + li];
      float cy2 = tile[3 * TC + li];
      float ca  = tile[4 * TC + li];
      float iw = fmaxf(fminf(rx2, cx2) - fmaxf(rx1, cx1) + 1.0f, 0.f);
      float ih = fmaxf(fminf(ry2, cy2) - fmaxf(ry1, cy1) + 1.0f, 0.f);
      bool p = (j < K) && (j != i) && (iw * ih >= NMS_T * ca);   // ov = iw*ih/area[j] >= 0.7
      u32 word = __builtin_amdgcn_ballot_w32(p);
      int gw = (c0 >> 5) + widx;
      if (lane == 0 && gw < NW) mask[(size_t)i * NW + gw] = word;
    }
    __syncthreads();
  }
}

// ---- stage 7: serial greedy suppression scan; single wave, valid mask in VGPRs ----
__global__ __launch_bounds__(32) void nms_reduce(const u32* __restrict__ mask,
                                                 u32* __restrict__ validOut) {
  int lane = threadIdx.x;                           // 32 lanes; lane l owns words l*6 .. l*6+5
  u32 v0, v1, v2, v3, v4, v5;
  {
    u32 vv[6];
    for (int q = 0; q < 6; ++q) {
      int w = lane * 6 + q;
      u32 init = 0u;
      if (w < NW) init = 0xFFFFFFFFu;
      if (w == NW - 1) init = (1u << (K - (NW - 1) * 32)) - 1u;  // last word: 16 valid bits
      vv[q] = init;
    }
    v0 = vv[0]; v1 = vv[1]; v2 = vv[2]; v3 = vv[3]; v4 = vv[4]; v5 = vv[5];
  }
  for (int i = 0; i < K; ++i) {
    int w = i >> 5, b = i & 31;
    int owner = w / 6, q = w % 6;                   // uniform across lanes
    u32 my;
    switch (q) {
      case 0: my = v0; break; case 1: my = v1; break; case 2: my = v2; break;
      case 3: my = v3; break; case 4: my = v4; break; default: my = v5; break;
    }
    u32 word = (u32)__shfl((int)my, owner, 32);
    if ((word >> b) & 1u) {                         // row i still valid -> suppress its overlaps
      const u32* row = mask + (size_t)i * NW;
      int w0 = lane * 6;
      if (w0 + 0 < NW) v0 &= ~row[w0 + 0];
      if (w0 + 1 < NW) v1 &= ~row[w0 + 1];
      if (w0 + 2 < NW) v2 &= ~row[w0 + 2];
      if (w0 + 3 < NW) v3 &= ~row[w0 + 3];
      if (w0 + 4 < NW) v4 &= ~row[w0 + 4];
      if (w0 + 5 < NW) v5 &= ~row[w0 + 5];
    }
  }
  int w0 = lane * 6;
  if (w0 + 0 < NW) validOut[w0 + 0] = v0;
  if (w0 + 1 < NW) validOut[w0 + 1] = v1;
  if (w0 + 2 < NW) validOut[w0 + 2] = v2;
  if (w0 + 3 < NW) validOut[w0 + 3] = v3;
  if (w0 + 4 < NW) validOut[w0 + 4] = v4;
  if (w0 + 5 < NW) validOut[w0 + 5] = v5;
}

// ---- stage 8a: exclusive prefix popcount over valid words ----
__global__ __launch_bounds__(256) void prefix_kernel(const u32* __restrict__ valid,
                                                     u32* __restrict__ prefix) {
  __shared__ u32 pc[NW];
  int t = threadIdx.x;
  if (t < NW) pc[t] = (u32)__popc(valid[t]);
  __syncthreads();
  if (t == 0) {
    u32 s = 0;
    for (int w = 0; w < NW; ++w) { prefix[w] = s; s += pc[w]; }
  }
}

__global__ __launch_bounds__(256) void zero_out(float* __restrict__ out, int n) {
  int i = blockIdx.x * blockDim.x + threadIdx.x;
  if (i < n) out[i] = 0.f;
}

// ---- stage 8b: scatter first 300 surviving boxes (sorted order) to output ----
__global__ __launch_bounds__(256) void scatter_out(const float4* __restrict__ sprops,
                                                   const u32* __restrict__ valid,
                                                   const u32* __restrict__ prefix,
                                                   float4* __restrict__ out) {
  int t = blockIdx.x * blockDim.x + threadIdx.x;
  if (t >= K) return;
  int wi = t >> 5, b = t & 31;
  u32 v = valid[wi];
  if ((v >> b) & 1u) {
    u32 rank = prefix[wi] + (u32)__popc(v & ((1u << b) - 1u));
    if (rank < 300u) out[rank] = sprops[t];
  }
}

extern "C" void kernel_launch(void* const* d_in, const int* in_sizes, int n_in,
                              void* d_out, int out_size, void* d_ws, size_t ws_size,
                              hipStream_t stream) {
  const float* scores = (const float*)d_in[0];
  const float* deltas = (const float*)d_in[1];
  float* out = (float*)d_out;
  char* ws = (char*)d_ws;

  float4* boxes  = (float4*)(ws + OFF_BOXES);
  u64*    keys1  = (u64*)   (ws + OFF_KEYS1);
  float4* props  = (float4*)(ws + OFF_PROPS);
  u64*    keys2  = (u64*)   (ws + OFF_KEYS2);
  float4* sprops = (float4*)(ws + OFF_SPROPS);
  float*  cols   = (float*) (ws + OFF_COLS);
  u32*    maskp  = (u32*)   (ws + OFF_MASK);
  u32*    validp = (u32*)   (ws + OFF_VALID);
  u32*    prefp  = (u32*)   (ws + OFF_PREFIX);

  build_boxes<<<NPAD1 / 256, 256, 0, stream>>>(scores, deltas, boxes, keys1);

  // sort 1: 2^18 keys, LDS-fused bitonic (k<=512 fused; per-k tails j<=256 fused)
  bitonic_local_full<<<NPAD1 / 512, 256, 0, stream>>>(keys1);
  for (u32 k = 1024; k <= NPAD1; k <<= 1) {
    for (u32 j = k >> 1; j >= 512; j >>= 1)
      bitonic_step<<<NPAD1 / 256, 256, 0, stream>>>(keys1, j, k, NPAD1);
    bitonic_local<<<NPAD1 / 512, 256, 0, stream>>>(keys1, k);
  }

  gather_props<<<NPAD2 / 256, 256, 0, stream>>>(keys1, boxes, props, keys2);

  // sort 2: 2^13 keys
  bitonic_local_full<<<NPAD2 / 512, 256, 0, stream>>>(keys2);
  for (u32 k = 1024; k <= NPAD2; k <<= 1) {
    for (u32 j = k >> 1; j >= 512; j >>= 1)
      bitonic_step<<<NPAD2 / 256, 256, 0, stream>>>(keys2, j, k, NPAD2);
    bitonic_local<<<NPAD2 / 512, 256, 0, stream>>>(keys2, k);
  }

  build_cols<<<(K + 255) / 256, 256, 0, stream>>>(keys2, props, sprops, cols);

  nms_mask<<<K, 256, 0, stream>>>(cols, maskp);

  nms_reduce<<<1, 32, 0, stream>>>(maskp, validp);

  prefix_kernel<<<1, 256, 0, stream>>>(validp, prefp);

  zero_out<<<(1200 + 255) / 256, 256, 0, stream>>>(out, 1200);

  scatter_out<<<(K + 255) / 256, 256, 0, stream>>>(sprops, validp, prefp, (float4*)out);
}